// RCP_LO_49246095016585
// MI455X (gfx1250) — compile-verified
//
#include <hip/hip_runtime.h>
#include <hip/hip_bf16.h>

#define NEI 32
#define WN  8

typedef __attribute__((ext_vector_type(2))) float v2f;
typedef __attribute__((ext_vector_type(8))) float v8f;

__device__ __forceinline__ float lrelu(float v, float s) { return v > 0.f ? v : v * s; }

// ---------------------------------------------------------------------------
// Generic fused GEMM: Y[rows,N] = leaky_relu(X[rows,K] * W[N,K]^T + bias, slope)
// One wave per 16x16 output tile using V_WMMA_F32_16X16X4_F32 (fp32 WMMA).
// A frag (16x4 f32): lanes 0-15 hold M=0..15; VGPR0 = K=kb, VGPR1 = K=kb+1,
//                    kb = 0 for lanes 0-15, 2 for lanes 16-31.
// B frag (4x16 f32): mirrored — lane&15 selects N, same kb split for K.
// C/D (16x16 f32):   VGPR r, lane l -> row = r + 8*(l>=16), col = l&15.
// ---------------------------------------------------------------------------
__global__ void gemm_lrelu_f32wmma(const float* __restrict__ X,
                                   const float* __restrict__ W,
                                   const float* __restrict__ bias,
                                   float* __restrict__ Y,
                                   int rows, int K, int N, float slope)
{
  const int lane = threadIdx.x & 31;
  const int wave = blockIdx.x * (blockDim.x >> 5) + (threadIdx.x >> 5);
  const int colTiles = N >> 4;
  const int rowTiles = (rows + 15) >> 4;
  if (wave >= rowTiles * colTiles) return;
  const int tm   = wave / colTiles;
  const int tn   = wave - tm * colTiles;
  const int row0 = tm << 4;
  const int col0 = tn << 4;

  const int mn   = lane & 15;        // M index for A, N index for B
  const int half = lane >> 4;
  const int kb   = half << 1;        // 0 or 2

  const int rA = min(row0 + mn, rows - 1);      // clamp pad rows (stores guarded)
  const float* __restrict__ xrow = X + (size_t)rA * K;
  const float* __restrict__ wrow = W + (size_t)(col0 + mn) * K;

  __builtin_prefetch(xrow, 0, 0);               // gfx1250 global_prefetch_b8
  __builtin_prefetch(wrow, 0, 0);

  v8f acc = {};
  int k = 0;
  for (; k + 4 <= K; k += 4) {
    v2f a, b;
    a.x = xrow[k + kb];
    a.y = xrow[k + kb + 1];
    b.x = wrow[k + kb];
    b.y = wrow[k + kb + 1];
    acc = __builtin_amdgcn_wmma_f32_16x16x4_f32(false, a, false, b,
                                                (short)0, acc, false, false);
  }
  if (k < K) {                                   // remainder (e.g. K == 3)
    v2f a = {}, b = {};
    if (k + kb < K)     { a.x = xrow[k + kb];     b.x = wrow[k + kb]; }
    if (k + kb + 1 < K) { a.y = xrow[k + kb + 1]; b.y = wrow[k + kb + 1]; }
    acc = __builtin_amdgcn_wmma_f32_16x16x4_f32(false, a, false, b,
                                                (short)0, acc, false, false);
  }

  const float bv = bias ? bias[col0 + mn] : 0.f;
#pragma unroll
  for (int r = 0; r < 8; ++r) {
    const int rr = row0 + r + (half << 3);
    if (rr < rows) {
      Y[(size_t)rr * N + col0 + mn] = lrelu(acc[r] + bv, slope);
    }
  }
}

// ---------------------------------------------------------------------------
// KNN: for each query, 32 nearest refs (ascending distance, lowest-index ties).
// Thread-per-query with block-shared ref tiles in LDS.
// ---------------------------------------------------------------------------
__global__ void knn_kernel(const float* __restrict__ q,    // [B,M,3]
                           const float* __restrict__ ref,  // [B,N,3]
                           int* __restrict__ idx,          // [B,M,NEI]
                           int M, int N)
{
  const int b  = blockIdx.y;
  const int mq = blockIdx.x * blockDim.x + threadIdx.x;
  const bool active = (mq < M);
  float qx = 0.f, qy = 0.f, qz = 0.f;
  if (active) {
    const float* p = q + ((size_t)b * M + mq) * 3;
    qx = p[0]; qy = p[1]; qz = p[2];
  }
  float bd[NEI]; int bi[NEI];
#pragma unroll
  for (int i = 0; i < NEI; ++i) { bd[i] = 3.0e38f; bi[i] = 0; }

  __shared__ float tile[256 * 3];
  const float* rbase = ref + (size_t)b * N * 3;
  for (int t0 = 0; t0 < N; t0 += 256) {
    const int cnt = min(256, N - t0);
    __syncthreads();
    for (int i = threadIdx.x; i < cnt * 3; i += blockDim.x)
      tile[i] = rbase[(size_t)t0 * 3 + i];
    __syncthreads();
    if (active) {
      for (int j = 0; j < cnt; ++j) {
        const float dx = tile[j * 3 + 0] - qx;
        const float dy = tile[j * 3 + 1] - qy;
        const float dz = tile[j * 3 + 2] - qz;
        const float d = dx * dx + dy * dy + dz * dz;
        if (d < bd[NEI - 1]) {
          int pos = NEI - 1;
          while (pos > 0 && bd[pos - 1] > d) {
            bd[pos] = bd[pos - 1]; bi[pos] = bi[pos - 1]; --pos;
          }
          bd[pos] = d; bi[pos] = t0 + j;
        }
      }
    }
  }
  if (active) {
    int* o = idx + ((size_t)b * M + mq) * NEI;
#pragma unroll
    for (int i = 0; i < NEI; ++i) o[i] = bi[i];
  }
}

// ---------------------------------------------------------------------------
// Farthest point sampling. One block per batch; dist[] lives in LDS.
// Emits indices (first = 0) and gathered coordinates.
// ---------------------------------------------------------------------------
__global__ void fps_kernel(const float* __restrict__ xyz,  // [B,N,3]
                           int N, int npoint,
                           int* __restrict__ outIdx,       // [B,npoint]
                           float* __restrict__ outPts)     // [B,npoint,3]
{
  const int b = blockIdx.x;
  const int tid = threadIdx.x;
  const int nthr = blockDim.x;
  extern __shared__ float dist[];          // N floats
  __shared__ float bd_s[256];
  __shared__ int   bi_s[256];
  __shared__ int   s_last;

  const float* base = xyz + (size_t)b * N * 3;
  for (int i = tid; i < N; i += nthr) dist[i] = 1.0e10f;
  if (tid == 0) s_last = 0;
  __syncthreads();

  for (int s = 0; s < npoint; ++s) {
    const int last = s_last;
    if (tid == 0) {
      outIdx[(size_t)b * npoint + s] = last;
      outPts[((size_t)b * npoint + s) * 3 + 0] = base[last * 3 + 0];
      outPts[((size_t)b * npoint + s) * 3 + 1] = base[last * 3 + 1];
      outPts[((size_t)b * npoint + s) * 3 + 2] = base[last * 3 + 2];
    }
    const float px = base[last * 3 + 0];
    const float py = base[last * 3 + 1];
    const float pz = base[last * 3 + 2];
    float lb = -1.f; int li = 0x7fffffff;
    for (int i = tid; i < N; i += nthr) {
      const float dx = base[i * 3 + 0] - px;
      const float dy = base[i * 3 + 1] - py;
      const float dz = base[i * 3 + 2] - pz;
      const float d = dx * dx + dy * dy + dz * dz;
      const float nd = fminf(dist[i], d);
      dist[i] = nd;
      if (nd > lb || (nd == lb && i < li)) { lb = nd; li = i; }
    }
    bd_s[tid] = lb; bi_s[tid] = li;
    __syncthreads();
    for (int off = nthr >> 1; off > 0; off >>= 1) {
      if (tid < off) {
        const float od = bd_s[tid + off]; const int oi = bi_s[tid + off];
        if (od > bd_s[tid] || (od == bd_s[tid] && oi < bi_s[tid])) {
          bd_s[tid] = od; bi_s[tid] = oi;
        }
      }
      __syncthreads();
    }
    if (tid == 0) s_last = bi_s[0];
    __syncthreads();
  }
}

// ---------------------------------------------------------------------------
// PointConv front-end: gather, relative coords, WeightNet MLP, k-aggregation.
// One 64-thread block per (b, query). Output agg layout: [B,M,(C+3)*WN],
// channel-major (c*WN + w) to match the reference reshape.
// ---------------------------------------------------------------------------
__global__ void pointconv_front(const float* __restrict__ feats,  // [B,Nref,C]
                                const float* __restrict__ refxyz, // [B,Nref,3]
                                const float* __restrict__ qxyz,   // [B,M,3]
                                const int* __restrict__ knn,      // [B,M,NEI]
                                const float* __restrict__ w1, const float* __restrict__ b1,
                                const float* __restrict__ w2, const float* __restrict__ b2,
                                const float* __restrict__ w3, const float* __restrict__ b3,
                                float* __restrict__ agg,
                                int M, int Nref, int C)
{
  const int b  = blockIdx.y;
  const int mq = blockIdx.x;
  const int tid = threadIdx.x;

  __shared__ float wgt[NEI][WN];
  __shared__ float nrm[NEI][3];
  __shared__ int   sidx[NEI];

  const int* kn = knn + ((size_t)b * M + mq) * NEI;
  if (tid < NEI) {
    const int id = kn[tid];
    sidx[tid] = id;
    const float* rp = refxyz + ((size_t)b * Nref + id) * 3;
    const float* qp = qxyz + ((size_t)b * M + mq) * 3;
    const float nx = rp[0] - qp[0], ny = rp[1] - qp[1], nz = rp[2] - qp[2];
    nrm[tid][0] = nx; nrm[tid][1] = ny; nrm[tid][2] = nz;
    float h1[WN], h2[WN];
#pragma unroll
    for (int o = 0; o < WN; ++o) {
      const float s = b1[o] + w1[o * 3 + 0] * nx + w1[o * 3 + 1] * ny + w1[o * 3 + 2] * nz;
      h1[o] = s > 0.f ? s : 0.f;
    }
#pragma unroll
    for (int o = 0; o < WN; ++o) {
      float s = b2[o];
#pragma unroll
      for (int i = 0; i < WN; ++i) s += w2[o * WN + i] * h1[i];
      h2[o] = s > 0.f ? s : 0.f;
    }
#pragma unroll
    for (int o = 0; o < WN; ++o) {
      float s = b3[o];
#pragma unroll
      for (int i = 0; i < WN; ++i) s += w3[o * WN + i] * h2[i];
      wgt[tid][o] = s > 0.f ? s : 0.f;
    }
  }
  __syncthreads();

  const float* fb = feats + (size_t)b * Nref * C;
  float* out = agg + ((size_t)b * M + mq) * (size_t)(C + 3) * WN;
  for (int c = tid; c < C + 3; c += blockDim.x) {
    float acc[WN] = {};
    for (int k = 0; k < NEI; ++k) {
      const float f = (c < C) ? fb[(size_t)sidx[k] * C + c] : nrm[k][c - C];
#pragma unroll
      for (int w = 0; w < WN; ++w) acc[w] += f * wgt[k][w];
    }
#pragma unroll
    for (int w = 0; w < WN; ++w) out[c * WN + w] = acc[w];
  }
}

// ---------------------------------------------------------------------------
// Layout helpers
// ---------------------------------------------------------------------------
__global__ void t_b3n_to_bn3(const float* __restrict__ in, float* __restrict__ out, int N)
{
  const int b = blockIdx.y;
  const int i = blockIdx.x * blockDim.x + threadIdx.x;
  if (i < N) {
#pragma unroll
    for (int c = 0; c < 3; ++c)
      out[((size_t)b * N + i) * 3 + c] = in[((size_t)b * 3 + c) * N + i];
  }
}

__global__ void t_bnc_to_bcn(const float* __restrict__ in, float* __restrict__ out,
                             int Npts, int C)
{
  const int b = blockIdx.y;
  const int i = blockIdx.x * blockDim.x + threadIdx.x;
  if (i < Npts * C) {
    const int n = i / C, c = i - n * C;
    out[((size_t)b * C + c) * Npts + n] = in[((size_t)b * Npts + n) * C + c];
  }
}

__global__ void idx_to_f32(const int* __restrict__ in, float* __restrict__ out, int n)
{
  const int i = blockIdx.x * blockDim.x + threadIdx.x;
  if (i < n) out[i] = (float)in[i];
}

// ---------------------------------------------------------------------------
// Host orchestration
// ---------------------------------------------------------------------------
extern "C" void kernel_launch(void* const* d_in, const int* in_sizes, int n_in,
                              void* d_out, int out_size, void* d_ws, size_t ws_size,
                              hipStream_t stream)
{
  const int B = 2, N0 = 8192;
  const int M1 = 2048, M2 = 512, M3 = 256, M4 = 64;

  const float* xyz_in   = (const float*)d_in[0];   // [B,3,N0]
  const float* color_in = (const float*)d_in[1];   // [B,3,N0]
  auto P = [&](int i) { return (const float*)d_in[i]; };
  // param slots (depth-first dict order)
  // 2,3: lift W,b | 4-9: L0 wn | 10,11: L0 lin | 12,13: level0_1
  // 14-19: L1 wn | 20,21: L1 lin | 22,23: level1_0 | 24,25: level1_1
  // 26-31: L2 wn | 32,33: L2 lin | 34,35: level2_0 | 36,37: level2_1
  // 38-43: L3 wn | 44,45: L3 lin | 46,47: level3_0 | 48,49: level3_1
  // 50-55: L4 wn | 56,57: L4 lin

  // ---- workspace carve-up (floats) ----
  float* ws = (float*)d_ws;
  size_t off = 0;
  float* XYZT = ws + off; off += (size_t)B * N0 * 3;       // [B,N0,3]
  float* COLT = ws + off; off += (size_t)B * N0 * 3;
  float* FA   = ws + off; off += (size_t)B * N0 * 64;      // max feature buffer
  float* FB   = ws + off; off += (size_t)B * N0 * 64;
  float* AGG  = ws + off; off += (size_t)B * N0 * 280;     // max agg buffer
  float* PC1  = ws + off; off += (size_t)B * M1 * 3;
  float* PC2  = ws + off; off += (size_t)B * M2 * 3;
  float* PC3  = ws + off; off += (size_t)B * M3 * 3;
  float* PC4  = ws + off; off += (size_t)B * M4 * 3;
  int* IDX = (int*)(ws + off); off += (size_t)B * N0 * NEI;
  int* I1  = (int*)(ws + off); off += (size_t)B * M1;
  int* I2  = (int*)(ws + off); off += (size_t)B * M2;
  int* I3  = (int*)(ws + off); off += (size_t)B * M3;
  int* I4  = (int*)(ws + off); off += (size_t)B * M4;

  // ---- output offsets (floats) ----
  float* out = (float*)d_out;
  size_t o = 0;
  float* o_xyz = out + o; o += (size_t)B * 3 * N0;
  float* o_pc1 = out + o; o += (size_t)B * 3 * M1;
  float* o_pc2 = out + o; o += (size_t)B * 3 * M2;
  float* o_pc3 = out + o; o += (size_t)B * 3 * M3;
  float* o_pc4 = out + o; o += (size_t)B * 3 * M4;
  float* o_f0  = out + o; o += (size_t)B * 32 * N0;
  float* o_f1  = out + o; o += (size_t)B * 64 * M1;
  float* o_f2  = out + o; o += (size_t)B * 128 * M2;
  float* o_f3  = out + o; o += (size_t)B * 256 * M3;
  float* o_f4  = out + o; o += (size_t)B * 512 * M4;
  float* o_i1  = out + o; o += (size_t)B * M1;
  float* o_i2  = out + o; o += (size_t)B * M2;
  float* o_i3  = out + o; o += (size_t)B * M3;
  float* o_i4  = out + o; o += (size_t)B * M4;

  auto gemm = [&](const float* X, const float* W, const float* bias, float* Y,
                  int rows, int K, int Nc) {
    const int waves = ((rows + 15) / 16) * (Nc / 16);
    const int blocks = (waves + 3) / 4;
    gemm_lrelu_f32wmma<<<dim3(blocks), dim3(128), 0, stream>>>(X, W, bias, Y,
                                                               rows, K, Nc, 0.1f);
  };
  auto knn = [&](const float* q, const float* r, int M, int N) {
    knn_kernel<<<dim3((M + 255) / 256, B), dim3(256), 0, stream>>>(q, r, IDX, M, N);
  };
  auto fps = [&](const float* pts, int N, int np, int* oi, float* op) {
    fps_kernel<<<dim3(B), dim3(256), (size_t)N * sizeof(float), stream>>>(pts, N, np, oi, op);
  };
  auto front = [&](const float* feats, const float* rxyz, const float* qxyz,
                   int wnBase, float* aggp, int M, int N, int C) {
    pointconv_front<<<dim3(M, B), dim3(64), 0, stream>>>(
        feats, rxyz, qxyz, IDX,
        P(wnBase + 0), P(wnBase + 1), P(wnBase + 2), P(wnBase + 3),
        P(wnBase + 4), P(wnBase + 5), aggp, M, N, C);
  };
  auto featOut = [&](const float* f, float* dst, int Npts, int C) {
    t_bnc_to_bcn<<<dim3((Npts * C + 255) / 256, B), dim3(256), 0, stream>>>(f, dst, Npts, C);
  };

  // ---- level 0 ----
  t_b3n_to_bn3<<<dim3((N0 + 255) / 256, B), dim3(256), 0, stream>>>(xyz_in, XYZT, N0);
  t_b3n_to_bn3<<<dim3((N0 + 255) / 256, B), dim3(256), 0, stream>>>(color_in, COLT, N0);
  gemm(COLT, P(2), P(3), FA, B * N0, 3, 32);                 // lift -> f [B,N0,32]
  knn(XYZT, XYZT, N0, N0);
  front(FA, XYZT, XYZT, 4, AGG, N0, N0, 32);                 // agg [B,N0,280]
  gemm(AGG, P(10), P(11), FB, B * N0, 35 * WN, 32);          // f0
  featOut(FB, o_f0, N0, 32);
  gemm(FB, P(12), P(13), FA, B * N0, 32, 64);                // f0_1

  // ---- level 1 ----
  fps(XYZT, N0, M1, I1, PC1);
  knn(PC1, XYZT, M1, N0);
  front(FA, XYZT, PC1, 14, AGG, M1, N0, 64);                 // agg [B,M1,536]
  gemm(AGG, P(20), P(21), FB, B * M1, 67 * WN, 64);          // f1a
  gemm(FB, P(22), P(23), FA, B * M1, 64, 64);                // f1
  featOut(FA, o_f1, M1, 64);
  gemm(FA, P(24), P(25), FB, B * M1, 64, 128);               // f1_2

  // ---- level 2 ----
  fps(PC1, M1, M2, I2, PC2);
  knn(PC2, PC1, M2, M1);
  front(FB, PC1, PC2, 26, AGG, M2, M1, 128);                 // agg [B,M2,1048]
  gemm(AGG, P(32), P(33), FA, B * M2, 131 * WN, 128);        // f2a
  gemm(FA, P(34), P(35), FB, B * M2, 128, 128);              // f2
  featOut(FB, o_f2, M2, 128);
  gemm(FB, P(36), P(37), FA, B * M2, 128, 256);              // f2_3

  // ---- level 3 ----
  fps(PC2, M2, M3, I3, PC3);
  knn(PC3, PC2, M3, M2);
  front(FA, PC2, PC3, 38, AGG, M3, M2, 256);                 // agg [B,M3,2072]
  gemm(AGG, P(44), P(45), FB, B * M3, 259 * WN, 256);        // f3a
  gemm(FB, P(46), P(47), FA, B * M3, 256, 256);              // f3
  featOut(FA, o_f3, M3, 256);
  gemm(FA, P(48), P(49), FB, B * M3, 256, 512);              // f3_4

  // ---- level 4 ----
  fps(PC3, M3, M4, I4, PC4);
  knn(PC4, PC3, M4, M3);
  front(FB, PC3, PC4, 50, AGG, M4, M3, 512);                 // agg [B,M4,4120]
  gemm(AGG, P(56), P(57), FA, B * M4, 515 * WN, 512);        // f4
  featOut(FA, o_f4, M4, 512);

  // ---- remaining outputs ----
  hipMemcpyAsync(o_xyz, xyz_in, (size_t)B * 3 * N0 * sizeof(float),
                 hipMemcpyDeviceToDevice, stream);
  featOut(PC1, o_pc1, M1, 3);
  featOut(PC2, o_pc2, M2, 3);
  featOut(PC3, o_pc3, M3, 3);
  featOut(PC4, o_pc4, M4, 3);
  idx_to_f32<<<dim3((B * M1 + 255) / 256), dim3(256), 0, stream>>>(I1, o_i1, B * M1);
  idx_to_f32<<<dim3((B * M2 + 255) / 256), dim3(256), 0, stream>>>(I2, o_i2, B * M2);
  idx_to_f32<<<dim3((B * M3 + 255) / 256), dim3(256), 0, stream>>>(I3, o_i3, B * M3);
  idx_to_f32<<<dim3((B * M4 + 255) / 256), dim3(256), 0, stream>>>(I4, o_i4, B * M4);
}